// GCNConv_S_86148454023368
// MI455X (gfx1250) — compile-verified
//
#include <hip/hip_runtime.h>
#include <stdint.h>

#define D_FEAT 128
#define TILE   1024          // edges staged per block
#define BLK    256           // 8 wave32 per block

// ---- ordered-int encoding for float atomicMax (monotone, self-inverse) ----
__device__ __forceinline__ int enc_f32(float f) {
    int i = __float_as_int(f);
    return i >= 0 ? i : (i ^ 0x7fffffff);
}
__device__ __forceinline__ float dec_f32(int i) {
    return __int_as_float(i >= 0 ? i : (i ^ 0x7fffffff));
}

// ---- async global->LDS staging of an edge-index tile (CDNA5 ASYNCcnt path) ----
__device__ __forceinline__ void async_stage_edges(const int* row, const int* col,
                                                  int* tile_rc, int base, int E) {
    const unsigned long long rptr = (unsigned long long)row;
    const unsigned long long cptr = (unsigned long long)col;
    #pragma unroll
    for (int t = threadIdx.x; t < TILE; t += BLK) {
        int g = base + t;
        if (g >= E) g = E - 1;                    // clamp tail (compute side re-checks)
        unsigned goff  = (unsigned)g << 2;        // byte offset
        unsigned lds_r = (unsigned)(uintptr_t)(&tile_rc[t]);
        unsigned lds_c = (unsigned)(uintptr_t)(&tile_rc[TILE + t]);
        asm volatile("global_load_async_to_lds_b32 %0, %1, %2"
                     :: "v"(lds_r), "v"(goff), "s"(rptr) : "memory");
        asm volatile("global_load_async_to_lds_b32 %0, %1, %2"
                     :: "v"(lds_c), "v"(goff), "s"(cptr) : "memory");
    }
    asm volatile("s_wait_asynccnt 0" ::: "memory");
    __syncthreads();
}

// ---- 1) zero deg_div[N*D] (b128 stores), deg[N], reset xmax ----
__global__ void k_init(float4* __restrict__ deg_div4, unsigned* __restrict__ deg,
                       int* __restrict__ xmax, int total4, int n) {
    int i = blockIdx.x * blockDim.x + threadIdx.x;
    int stride = gridDim.x * blockDim.x;
    const float4 z = {0.0f, 0.0f, 0.0f, 0.0f};
    for (; i < total4; i += stride) {
        deg_div4[i] = z;
        if (i < n) deg[i] = 0u;
        if (i == 0) *xmax = (int)0x80000000;   // < any encoded float
    }
}

// ---- 2) degree over col ----
__global__ void k_deg(const int* __restrict__ col, unsigned* __restrict__ deg, int E) {
    int e = blockIdx.x * blockDim.x + threadIdx.x;
    if (e < E) atomicAdd(&deg[col[e]], 1u);
}

// ---- 3) dis = deg > 0 ? deg^-1/2 : 0 ----
__global__ void k_dis(const unsigned* __restrict__ deg, float* __restrict__ dis, int n) {
    int i = blockIdx.x * blockDim.x + threadIdx.x;
    if (i < n) {
        unsigned d = deg[i];
        dis[i] = d ? __builtin_amdgcn_rsqf((float)d) : 0.0f;
    }
}

// ---- 4+5 fused) global max of x AND out = (1+eps)*x in one pass over x ----
__global__ void k_max_outinit(const float4* __restrict__ x4, const float* __restrict__ eps,
                              float4* __restrict__ out4, int n4, int* __restrict__ xmax) {
    const float e1 = 1.0f + eps[0];
    int i = blockIdx.x * blockDim.x + threadIdx.x;
    int stride = gridDim.x * blockDim.x;
    float m = -3.402823466e38f;
    for (; i < n4; i += stride) {
        float4 v = x4[i];
        m = fmaxf(m, fmaxf(fmaxf(v.x, v.y), fmaxf(v.z, v.w)));
        v.x *= e1; v.y *= e1; v.z *= e1; v.w *= e1;
        out4[i] = v;
    }
    #pragma unroll
    for (int off = 16; off; off >>= 1)
        m = fmaxf(m, __shfl_xor(m, off, 32));
    if ((threadIdx.x & 31) == 0) atomicMax(xmax, enc_f32(m));
}

// ---- 6) deg_div[row] += exp(pp*x[col]-M) * norm   (wave32 per edge) ----
__global__ void __launch_bounds__(BLK)
k_pass1(const int* __restrict__ row, const int* __restrict__ col,
        const float* __restrict__ x, const float* __restrict__ dis,
        const float* __restrict__ p, const int* __restrict__ xmax,
        float* __restrict__ deg_div, int E) {
    __shared__ int tile_rc[2 * TILE];
    const int base = blockIdx.x * TILE;
    async_stage_edges(row, col, tile_rc, base, E);

    const float pv = p[0];
    const float pp = 2.0f / (1.0f + __expf(-pv));
    const float M  = pp * dec_f32(*xmax);

    const int lane = threadIdx.x & 31;
    const int f    = lane << 2;                   // float4 slice of 128 feats
    const int kBeg = (threadIdx.x >> 5) * (TILE / (BLK / 32));
    const int kEnd = kBeg + (TILE / (BLK / 32));
    for (int k = kBeg; k < kEnd; ++k) {
        int e = base + k;
        if (e >= E) break;
        int r = __builtin_amdgcn_readfirstlane(tile_rc[k]);
        int c = __builtin_amdgcn_readfirstlane(tile_rc[TILE + k]);
        float nrm = dis[r] * dis[c];
        if (nrm == 0.0f) continue;                // zero contribution, skip atomics
        const float4 xc = *(const float4*)(x + c * D_FEAT + f);
        float* dst = deg_div + r * D_FEAT + f;
        atomicAdd(dst + 0, __expf(pp * xc.x - M) * nrm);
        atomicAdd(dst + 1, __expf(pp * xc.y - M) * nrm);
        atomicAdd(dst + 2, __expf(pp * xc.z - M) * nrm);
        atomicAdd(dst + 3, __expf(pp * xc.w - M) * nrm);
    }
}

// ---- 7) out[row] += exp(pp*x[col]-M)*x[col]*norm / (deg_div[row]+1e-6) ----
__global__ void __launch_bounds__(BLK)
k_pass2(const int* __restrict__ row, const int* __restrict__ col,
        const float* __restrict__ x, const float* __restrict__ dis,
        const float* __restrict__ p, const int* __restrict__ xmax,
        const float* __restrict__ deg_div, float* __restrict__ out, int E) {
    __shared__ int tile_rc[2 * TILE];
    const int base = blockIdx.x * TILE;
    async_stage_edges(row, col, tile_rc, base, E);

    const float pv = p[0];
    const float pp = 2.0f / (1.0f + __expf(-pv));
    const float M  = pp * dec_f32(*xmax);

    const int lane = threadIdx.x & 31;
    const int f    = lane << 2;
    const int kBeg = (threadIdx.x >> 5) * (TILE / (BLK / 32));
    const int kEnd = kBeg + (TILE / (BLK / 32));
    for (int k = kBeg; k < kEnd; ++k) {
        int e = base + k;
        if (e >= E) break;
        int r = __builtin_amdgcn_readfirstlane(tile_rc[k]);
        int c = __builtin_amdgcn_readfirstlane(tile_rc[TILE + k]);
        float nrm = dis[r] * dis[c];
        if (nrm == 0.0f) continue;
        const float4 xc = *(const float4*)(x + c * D_FEAT + f);
        const float4 dd = *(const float4*)(deg_div + r * D_FEAT + f);
        float* dst = out + r * D_FEAT + f;
        atomicAdd(dst + 0, __expf(pp * xc.x - M) * xc.x * nrm * __builtin_amdgcn_rcpf(dd.x + 1e-6f));
        atomicAdd(dst + 1, __expf(pp * xc.y - M) * xc.y * nrm * __builtin_amdgcn_rcpf(dd.y + 1e-6f));
        atomicAdd(dst + 2, __expf(pp * xc.z - M) * xc.z * nrm * __builtin_amdgcn_rcpf(dd.z + 1e-6f));
        atomicAdd(dst + 3, __expf(pp * xc.w - M) * xc.w * nrm * __builtin_amdgcn_rcpf(dd.w + 1e-6f));
    }
}

extern "C" void kernel_launch(void* const* d_in, const int* in_sizes, int n_in,
                              void* d_out, int out_size, void* d_ws, size_t ws_size,
                              hipStream_t stream) {
    (void)n_in; (void)out_size; (void)ws_size;
    const float* x   = (const float*)d_in[0];
    const int*   ei  = (const int*)d_in[1];
    const float* eps = (const float*)d_in[2];
    const float* p   = (const float*)d_in[3];

    const int ND = in_sizes[0];            // N * 128
    const int N  = ND / D_FEAT;
    const int E  = in_sizes[1] / 2;
    const int* row = ei;
    const int* col = ei + E;
    float* out = (float*)d_out;

    // workspace layout: deg[N] u32 | dis[N] f32 | xmax i32 | (256B align) | deg_div[N*D] f32
    char* ws = (char*)d_ws;
    unsigned* deg = (unsigned*)ws;
    float*    dis = (float*)(ws + (size_t)N * 4);
    int*     xmax = (int*)(ws + (size_t)N * 8);
    size_t dd_off = ((size_t)N * 8 + 4 + 255) & ~(size_t)255;
    float* deg_div = (float*)(ws + dd_off);

    const int n4 = ND / 4;
    const int nbE    = (E + BLK - 1) / BLK;
    const int nbTile = (E + TILE - 1) / TILE;

    k_init       <<<2048, BLK, 0, stream>>>((float4*)deg_div, deg, xmax, n4, N);
    k_deg        <<<nbE, BLK, 0, stream>>>(col, deg, E);
    k_dis        <<<(N + BLK - 1) / BLK, BLK, 0, stream>>>(deg, dis, N);
    k_max_outinit<<<1024, BLK, 0, stream>>>((const float4*)x, eps, (float4*)out, n4, xmax);
    k_pass1      <<<nbTile, BLK, 0, stream>>>(row, col, x, dis, p, xmax, deg_div, E);
    k_pass2      <<<nbTile, BLK, 0, stream>>>(row, col, x, dis, p, xmax, deg_div, out, E);
}